// VAELoss_18554258718750
// MI455X (gfx1250) — compile-verified
//
#include <hip/hip_runtime.h>
#include <hip/hip_bf16.h>

typedef __attribute__((ext_vector_type(2))) float v2f;
typedef __attribute__((ext_vector_type(4))) float v4f;
typedef __attribute__((ext_vector_type(8))) float v8f;

#define N_ROWS 4096
#define V_DIM  20000
#define LAT    32
#define N_USR  128
#define L_DIM  1024
#define BLK    256

// ---------------------------------------------------------------------------
// Block-wide f32 sum whose final stage runs on the CDNA5 matrix pipe:
// 256 partials -> wave 0 performs 4 chained V_WMMA_F32_16X16X4_F32 with an
// all-ones B matrix (D = A*1 + C == row sums, accumulated through C), then
// two cross-lane adds. Full f32 precision (required: hinge sums ~7e7).
// Result valid on threadIdx.x == 0.
// ---------------------------------------------------------------------------
__device__ __forceinline__ float block_sum_wmma(float v) {
    __shared__ float sb[BLK];
    sb[threadIdx.x] = v;
    __syncthreads();
    float result = 0.0f;
    if (threadIdx.x < 32) {               // wave 0, EXEC all-ones (wave32)
        const int lane = threadIdx.x;
        v8f acc = {};
        v2f b;  b.x = 1.0f; b.y = 1.0f;   // all-ones B (4x16)
#pragma unroll
        for (int tile = 0; tile < 4; ++tile) {
            v2f a;                         // 64 values per WMMA: any bijection
            a.x = sb[tile * 64 + lane];    // into the 16x4 A layout sums the same
            a.y = sb[tile * 64 + 32 + lane];
            acc = __builtin_amdgcn_wmma_f32_16x16x4_f32(
                false, a, false, b, (short)0, acc, false, false);
        }
        float s = acc[0] + acc[1] + acc[2] + acc[3] +
                  acc[4] + acc[5] + acc[6] + acc[7];
        // lanes 0-15 hold sum of rows 0-7, lanes 16-31 hold sum of rows 8-15
        result = __shfl(s, 0, 32) + __shfl(s, 16, 32);
    }
    __syncthreads();                       // sb reusable by next call
    return result;
}

// ---------------------------------------------------------------------------
// KLD partials: sum of zm^2 + zs^2 - log(zs^2) - 1 over 131072 elements.
// ---------------------------------------------------------------------------
__global__ void kld_partial_k(const float* __restrict__ zm,
                              const float* __restrict__ zs,
                              float* __restrict__ part, int n) {
    float acc = 0.0f;
    const int stride = gridDim.x * blockDim.x;
    for (int i = blockIdx.x * blockDim.x + threadIdx.x; i < n; i += stride) {
        float a  = zm[i];
        float s  = zs[i];
        float s2 = s * s;
        acc += fmaf(a, a, s2) - __logf(s2) - 1.0f;
    }
    float bs = block_sum_wmma(acc);
    if (threadIdx.x == 0) part[blockIdx.x] = bs;
}

// ---------------------------------------------------------------------------
// Cross entropy: one block per row. Single-pass online softmax over 20000
// floats using coalesced non-temporal float4 (B128) loads -> each of the
// 328 MB is read exactly once (bandwidth-bound, no L2 pollution).
// ce_row[row] = logsumexp(m[row,:]) - m[row, t[row]]
// ---------------------------------------------------------------------------
__global__ void ce_rows_k(const float* __restrict__ m,
                          const int* __restrict__ t,
                          float* __restrict__ ce_row) {
    const int row = blockIdx.x;
    const int tid = threadIdx.x;
    const float* p = m + (size_t)row * V_DIM;            // 80000 B stride, 16B aligned
    const v4f* p4 = (const v4f*)p;
    const int nvec = V_DIM / 4;                           // 5000

    float vmax = -__builtin_inff();
    float vsum = 0.0f;
    for (int i = tid; i < nvec; i += BLK) {
        v4f v = __builtin_nontemporal_load(&p4[i]);
        float cmax = fmaxf(fmaxf(v.x, v.y), fmaxf(v.z, v.w));
        if (cmax > vmax) { vsum *= __expf(vmax - cmax); vmax = cmax; }
        vsum += __expf(v.x - vmax) + __expf(v.y - vmax) +
                __expf(v.z - vmax) + __expf(v.w - vmax);
    }

    __shared__ float smax[BLK];
    __shared__ float ssum[BLK];
    smax[tid] = vmax;
    ssum[tid] = vsum;
    __syncthreads();
#pragma unroll
    for (int off = BLK / 2; off > 0; off >>= 1) {
        if (tid < off) {
            float m1 = smax[tid],       s1 = ssum[tid];
            float m2 = smax[tid + off], s2 = ssum[tid + off];
            float mm = fmaxf(m1, m2);
            smax[tid] = mm;
            ssum[tid] = s1 * __expf(m1 - mm) + s2 * __expf(m2 - mm);
        }
        __syncthreads();
    }
    if (tid == 0) {
        float lse = smax[0] + logf(ssum[0]);
        ce_row[row] = lse - p[t[row]];
    }
}

// ---------------------------------------------------------------------------
// Pairwise hinge: one block per user. Stage o/y rows (8 KB) in LDS, then
// sweep all i<j pairs. Per-user sum + valid count (exact in f32: < 2^24).
// ---------------------------------------------------------------------------
__global__ void pairwise_k(const float* __restrict__ o,
                           const int* __restrict__ y,
                           float* __restrict__ psum,
                           float* __restrict__ pcnt) {
    const int u = blockIdx.x;
    __shared__ float so[L_DIM];
    __shared__ int   sy[L_DIM];
    for (int i = threadIdx.x; i < L_DIM; i += BLK) {
        so[i] = o[(size_t)u * L_DIM + i];
        sy[i] = y[(size_t)u * L_DIM + i];
    }
    __syncthreads();

    float hs  = 0.0f;
    float cnt = 0.0f;
    for (int i = threadIdx.x; i < L_DIM - 1; i += BLK) {
        const float oi = so[i];
        const int   yi = sy[i];
        for (int j = i + 1; j < L_DIM; ++j) {
            int   yd = yi - sy[j];
            float sgn = (yd > 0) ? 1.0f : -1.0f;
            float h   = fmaxf(0.0f, 1.0f - sgn * (oi - so[j]));
            if (yd != 0) { hs += h; cnt += 1.0f; }
        }
    }
    float bs = block_sum_wmma(hs);
    float bc = block_sum_wmma(cnt);
    if (threadIdx.x == 0) { psum[u] = bs; pcnt[u] = bc; }
}

// ---------------------------------------------------------------------------
// Finalize: deterministic fixed-order reduction of all partials, combine.
// ---------------------------------------------------------------------------
__global__ void finalize_k(const float* __restrict__ kldp,   // [128]
                           const float* __restrict__ ce_row, // [4096]
                           const float* __restrict__ psum,   // [128]
                           const float* __restrict__ pcnt,   // [128]
                           float* __restrict__ out) {
    const int tid = threadIdx.x;

    float lce = 0.0f;
    for (int i = tid; i < N_ROWS; i += BLK) lce += ce_row[i];
    float ce_sum = block_sum_wmma(lce);

    float lk = (tid < 128) ? kldp[tid] : 0.0f;
    float kld_sum = block_sum_wmma(lk);

    float lp = (tid < 128) ? psum[tid] : 0.0f;
    float ps_sum = block_sum_wmma(lp);

    if (tid == 0) {
        double c = 0.0;
        for (int i = 0; i < 128; ++i) c += (double)pcnt[i];  // exact count
        if (c < 1.0) c = 1.0;
        float kld      = kld_sum / (float)(N_ROWS * LAT);
        float like     = ce_sum / (float)N_ROWS;
        float pairwise = (float)((double)ps_sum / c);
        out[0] = 0.05f * kld + 0.2f * like + 0.75f * pairwise;
    }
}

// ---------------------------------------------------------------------------
extern "C" void kernel_launch(void* const* d_in, const int* in_sizes, int n_in,
                              void* d_out, int out_size, void* d_ws, size_t ws_size,
                              hipStream_t stream) {
    const float* m  = (const float*)d_in[0];   // [4096, 20000]
    const float* zm = (const float*)d_in[1];   // [4096, 32]
    const float* zs = (const float*)d_in[2];   // [4096, 32]
    const float* o  = (const float*)d_in[3];   // [128, 1024]
    const int*   y  = (const int*)d_in[4];     // [128, 1024]
    const int*   t  = (const int*)d_in[5];     // [4096]

    float* ws     = (float*)d_ws;
    float* ce_row = ws;                        // 4096 floats
    float* kldp   = ws + N_ROWS;               // 128 floats
    float* psum   = ws + N_ROWS + 128;         // 128 floats
    float* pcnt   = ws + N_ROWS + 256;         // 128 floats

    kld_partial_k<<<128, BLK, 0, stream>>>(zm, zs, kldp, N_ROWS * LAT);
    ce_rows_k<<<N_ROWS, BLK, 0, stream>>>(m, t, ce_row);
    pairwise_k<<<N_USR, BLK, 0, stream>>>(o, y, psum, pcnt);
    finalize_k<<<1, BLK, 0, stream>>>(kldp, ce_row, psum, pcnt, (float*)d_out);
}